// dynemb_52089363366206
// MI455X (gfx1250) — compile-verified
//
#include <hip/hip_runtime.h>
#include <hip/hip_bf16.h>

typedef float v2f __attribute__((ext_vector_type(2)));
typedef float v8f __attribute__((ext_vector_type(8)));

#define EMDIM 128
#define KSTEPS (EMDIM / 4)   // 32 WMMA K-steps of 4

__device__ __forceinline__ float softplusf(float x) {
    // numerically stable softplus: max(x,0) + log1p(exp(-|x|))
    return fmaxf(x, 0.0f) + log1pf(expf(-fabsf(x)));
}

__global__ __launch_bounds__(128)
void dynemb_wmma_kernel(const int* __restrict__ input,
                        const int* __restrict__ neg_idx,
                        const float* __restrict__ emb,
                        const float* __restrict__ W0,
                        const float* __restrict__ b0p,
                        const float* __restrict__ W1,
                        const float* __restrict__ b1p,
                        const float* __restrict__ psi,
                        float* __restrict__ out,
                        int B, int NN, int K)
{
    const int b    = blockIdx.x;
    const int tid  = threadIdx.x;
    const int wave = tid >> 5;
    const int lane = tid & 31;
    const int col  = lane & 15;
    const int hi   = lane >> 4;

    __shared__ float red[4][4];     // per-wave partials of the 4 positive dots
    __shared__ float S[4][16][4];   // [wave][negative-in-tile][dot: Wa0,Wa1,Wb0,Wb1]
    __shared__ float sred[4];       // per-wave surv partials
    __shared__ float dots[4];       // e1.Wa0, e1.Wa1, e2.Wb0, e2.Wb1

    const int v1 = input[b * 5 + 0];
    const int v2 = input[b * 5 + 1];
    const int kb = input[b * 5 + 4] % K;

    // ---------------- positive-pair dots (blockDim == EMDIM == 128) -------
    {
        const int d = tid;
        const float e1d = emb[(size_t)v1 * EMDIM + d];
        const float e2d = emb[(size_t)v2 * EMDIM + d];
        float p0 = e1d * W0[d];          // e1 . Wa0
        float p1 = e1d * W1[d];          // e1 . Wa1
        float p2 = e2d * W0[EMDIM + d];  // e2 . Wb0
        float p3 = e2d * W1[EMDIM + d];  // e2 . Wb1
        #pragma unroll
        for (int off = 16; off >= 1; off >>= 1) {
            p0 += __shfl_xor(p0, off, 32);
            p1 += __shfl_xor(p1, off, 32);
            p2 += __shfl_xor(p2, off, 32);
            p3 += __shfl_xor(p3, off, 32);
        }
        if (lane == 0) {
            red[wave][0] = p0; red[wave][1] = p1;
            red[wave][2] = p2; red[wave][3] = p3;
        }
    }
    __syncthreads();
    if (tid < 4)   // fixed-order sum -> deterministic
        dots[tid] = red[0][tid] + red[1][tid] + red[2][tid] + red[3][tid];
    __syncthreads();

    // ---------------- negative gathers + WMMA f32 16x16x4 chains ----------
    // A fragment (16x4 f32): lane<16 holds row[4t],row[4t+1]; lane>=16 holds
    // row[4t+2],row[4t+3]  -> per-lane b64 load at float offset 4t + 2*hi.
    int n = wave * 16 + col;
    if (n >= NN) n = NN - 1;                       // safety clamp (NN==64 expected)
    const int i1 = neg_idx[((size_t)b * NN + n) * 2 + 0];
    const int i2 = neg_idx[((size_t)b * NN + n) * 2 + 1];
    const v2f* __restrict__ pa1 = (const v2f*)(emb + (size_t)i1 * EMDIM + hi * 2);
    const v2f* __restrict__ pa2 = (const v2f*)(emb + (size_t)i2 * EMDIM + hi * 2);

    // B fragment (4x16 f32): column 'col' of W; cols 0..3 = Wa0,Wa1,Wb0,Wb1;
    // cols 4..15 alias col 3 (valid address, results never read).
    const int c = (col < 4) ? col : 3;
    const v2f* __restrict__ pw =
        (const v2f*)(((c & 1) ? W1 : W0) + ((c >= 2) ? EMDIM : 0) + hi * 2);

    v8f acc1 = {0.f, 0.f, 0.f, 0.f, 0.f, 0.f, 0.f, 0.f};   // eo1 . W
    v8f acc2 = {0.f, 0.f, 0.f, 0.f, 0.f, 0.f, 0.f, 0.f};   // eo2 . W
    #pragma unroll
    for (int t = 0; t < KSTEPS; ++t) {
        const v2f a1 = pa1[2 * t];
        const v2f a2 = pa2[2 * t];
        const v2f bv = pw[2 * t];
        acc1 = __builtin_amdgcn_wmma_f32_16x16x4_f32(false, a1, false, bv,
                                                     (short)0, acc1, false, false);
        acc2 = __builtin_amdgcn_wmma_f32_16x16x4_f32(false, a2, false, bv,
                                                     (short)0, acc2, false, false);
    }

    // C layout: lane<16 vgpr j = C[M=j][N=col]; lane>=16 vgpr j = C[M=8+j][N=col]
    if (col < 2) {              // eo1 . Wa_k  (cols 0,1 of acc1)
        #pragma unroll
        for (int j = 0; j < 8; ++j) S[wave][hi * 8 + j][col] = acc1[j];
    } else if (col < 4) {       // eo2 . Wb_k  (cols 2,3 of acc2)
        #pragma unroll
        for (int j = 0; j < 8; ++j) S[wave][hi * 8 + j][col] = acc2[j];
    }
    __syncthreads();

    // ---------------- softplus + reductions -------------------------------
    const float bb0 = b0p[0], bb1 = b1p[0];
    const float ps0 = psi[0];
    const float ps1 = (K > 1) ? psi[1] : psi[0];
    const float a10 = dots[0], a11 = dots[1], a20 = dots[2], a21 = dots[3];

    float contrib = 0.0f;
    if (tid < NN) {
        const int w  = tid >> 4;
        const int mm = tid & 15;
        const float A10 = S[w][mm][0];   // eo1 . Wa0
        const float A11 = S[w][mm][1];   // eo1 . Wa1
        const float A20 = S[w][mm][2];   // eo2 . Wb0
        const float A21 = S[w][mm][3];   // eo2 . Wb1
        const float s1_0 = a10 + A20 + bb0;
        const float s1_1 = a11 + A21 + bb1;
        const float s2_0 = A10 + a20 + bb0;
        const float s2_1 = A11 + a21 + bb1;
        contrib = ps0 * softplusf(s1_0 / ps0) + ps1 * softplusf(s1_1 / ps1)
                + ps0 * softplusf(s2_0 / ps0) + ps1 * softplusf(s2_1 / ps1);
    }
    #pragma unroll
    for (int off = 16; off >= 1; off >>= 1)
        contrib += __shfl_xor(contrib, off, 32);
    if (lane == 0) sred[wave] = contrib;
    __syncthreads();

    if (tid == 0) {
        const float surv = (sred[0] + sred[1] + sred[2] + sred[3]) / (float)NN;
        const float sk = (kb == 0) ? (a10 + a20 + bb0) : (a11 + a21 + bb1);
        const float pk = (kb == 0) ? ps0 : ps1;
        out[b]     = pk * softplusf(sk / pk);   // intensity, shape (1,B)
        out[B + b] = surv;                      // surv, shape (1,B)
    }
}

extern "C" void kernel_launch(void* const* d_in, const int* in_sizes, int n_in,
                              void* d_out, int out_size, void* d_ws, size_t ws_size,
                              hipStream_t stream) {
    const int*   input   = (const int*)d_in[0];   // (B,5) int32
    const int*   neg_idx = (const int*)d_in[1];   // (B,NN,2) int32
    // d_in[2] = N (python scalar), value == NN; derive NN from sizes instead
    const float* emb     = (const float*)d_in[3]; // (NSIZE,128) f32
    const float* W0      = (const float*)d_in[4]; // (256,1) f32
    const float* b0      = (const float*)d_in[5]; // (1,) f32
    const float* W1      = (const float*)d_in[6]; // (256,1) f32
    const float* b1      = (const float*)d_in[7]; // (1,) f32
    const float* psi     = (const float*)d_in[8]; // (K,1) f32

    const int B  = in_sizes[0] / 5;
    const int NN = in_sizes[1] / (B * 2);
    const int K  = in_sizes[8];

    dynemb_wmma_kernel<<<B, 128, 0, stream>>>(input, neg_idx, emb, W0, b0, W1, b1,
                                              psi, (float*)d_out, B, NN, K);
}